// W_generation_67224828117241
// MI455X (gfx1250) — compile-verified
//
#include <hip/hip_runtime.h>
#include <hip/hip_bf16.h>

// Problem constants
#define B_SZ   4096
#define IN_SZ  1024
#define F_SZ   512
#define HID_SZ 128
#define D_SZ   128
#define RH_SZ  64
#define OUT_SZ 1
#define C_SZ   10
#define R_SZ   16
#define KU     1280          // C*D, flattened hyper contraction dim
#define O1     65536         // HID*F, fc1 output dim

// Workspace layout (float elements). Total ~3.18M floats (~12.7 MB).
enum : size_t {
  OFF_W     = 0,                              // (R, C)      = 160
  OFF_RAY2V = 256,                            // (R, D)      = 2048
  OFF_U     = 2304,                           // (R, C*D)    = 20480
  OFF_B1    = 22784,                          // (R, HID)    = 2048
  OFF_W2    = 24832,                          // (R, HID)    = 2048
  OFF_B2    = 26880,                          // (R,)        = 16
  OFF_W1    = 27136,                          // (R, O1)     = 1048576
  OFF_FEAT  = OFF_W1 + (size_t)R_SZ * O1,     // (B, F)      = 2097152
};

typedef __attribute__((ext_vector_type(2))) float v2f;
typedef __attribute__((ext_vector_type(4))) float v4f;
typedef __attribute__((ext_vector_type(8))) float v8f;

__device__ __forceinline__ v8f wmma_f32(v2f a, v2f b, v8f c) {
  // D = A(16x4,f32) x B(4x16,f32) + C(16x16,f32)
  return __builtin_amdgcn_wmma_f32_16x16x4_f32(false, a, false, b, (short)0, c,
                                               false, false);
}

// ---------------------------------------------------------------------------
// Kernel 1: tiny ray MLP. One block per ray r.
// Produces w (R,C), ray2v (R,D), U[r, c*D+d] = w[r,c]*ray2v[r,d].
// ---------------------------------------------------------------------------
__global__ void prep_kernel(const float* __restrict__ pref,
                            const float* __restrict__ wm1_w, const float* __restrict__ wm1_b,
                            const float* __restrict__ wm2_w, const float* __restrict__ wm2_b,
                            const float* __restrict__ ray0_w, const float* __restrict__ ray0_b,
                            const float* __restrict__ ray2_w, const float* __restrict__ ray2_b,
                            float* __restrict__ ws) {
  __shared__ float h_s[16];
  __shared__ float w_s[C_SZ];
  __shared__ float r1_s[RH_SZ];
  const int r = blockIdx.x;
  const int tid = threadIdx.x;
  const float p0 = pref[r * 2 + 0], p1 = pref[r * 2 + 1];

  if (tid < 16) {
    float v = p0 * wm1_w[tid * 2 + 0] + p1 * wm1_w[tid * 2 + 1] + wm1_b[tid];
    h_s[tid] = v > 0.f ? v : 0.f;
  }
  __syncthreads();
  if (tid < C_SZ) {
    float s = wm2_b[tid];
#pragma unroll
    for (int j = 0; j < 16; ++j) s += h_s[j] * wm2_w[tid * 16 + j];
    w_s[tid] = 1.f / (1.f + __expf(-s));
  }
  __syncthreads();
  if (tid == 0) {
    float t = 0.f;
    for (int c = 0; c < C_SZ; ++c) t += w_s[c];
    float inv = 1.f / t;
    for (int c = 0; c < C_SZ; ++c) w_s[c] *= inv;
  }
  __syncthreads();
  if (tid < C_SZ) ws[OFF_W + r * C_SZ + tid] = w_s[tid];

  if (tid < RH_SZ) {
    float m0 = 0.f, m1 = 0.f, bb = 0.f;
#pragma unroll
    for (int c = 0; c < C_SZ; ++c) {
      m0 += w_s[c] * ray0_w[c * RH_SZ * 2 + tid * 2 + 0];
      m1 += w_s[c] * ray0_w[c * RH_SZ * 2 + tid * 2 + 1];
      bb += w_s[c] * ray0_b[c * RH_SZ + tid];
    }
    float v = p0 * m0 + p1 * m1 + bb;
    r1_s[tid] = v > 0.f ? v : 0.f;
  }
  __syncthreads();
  if (tid < D_SZ) {
    float acc = 0.f;
#pragma unroll
    for (int c = 0; c < C_SZ; ++c) acc += w_s[c] * ray2_b[c * D_SZ + tid];
    for (int h = 0; h < RH_SZ; ++h) {
      float mw = 0.f;
#pragma unroll
      for (int c = 0; c < C_SZ; ++c)
        mw += w_s[c] * ray2_w[c * D_SZ * RH_SZ + tid * RH_SZ + h];
      acc += r1_s[h] * mw;
    }
    ws[OFF_RAY2V + r * D_SZ + tid] = acc;
#pragma unroll
    for (int c = 0; c < C_SZ; ++c)
      ws[OFF_U + r * KU + c * D_SZ + tid] = w_s[c] * acc;
  }
}

// ---------------------------------------------------------------------------
// Kernel 2: small hyper layers: out[r,o] = sum_{c,d} U[r,c*D+d]*Wc[c,o,d]
//                                         + sum_c w[r,c]*bc[c,o]
// ---------------------------------------------------------------------------
__global__ void small_hyper_kernel(const float* __restrict__ Wc,
                                   const float* __restrict__ bc,
                                   const float* __restrict__ ws,
                                   float* __restrict__ outp, int O) {
  const int r = blockIdx.x, o = threadIdx.x;
  if (o >= O) return;
  const float* U = ws + OFF_U + (size_t)r * KU;
  const float* w = ws + OFF_W + (size_t)r * C_SZ;
  float acc = 0.f;
  for (int c = 0; c < C_SZ; ++c) {
    const float* wrow = Wc + ((size_t)c * O + o) * D_SZ;
    const float* urow = U + c * D_SZ;
    float a2 = 0.f;
#pragma unroll 8
    for (int d = 0; d < D_SZ; ++d) a2 += urow[d] * wrow[d];
    acc += a2 + w[c] * bc[(size_t)c * O + o];
  }
  outp[(size_t)r * O + o] = acc;
}

// ---------------------------------------------------------------------------
// Kernel 3: W1 hyper GEMM via fp32 WMMA. M=16(R), N=65536, K=1280.
// The 335 MB fc1_w stream is the HBM-critical path. 256-thread blocks
// (8 waves, 256 o-columns each) stage 256x32 f32 chunks (32 KB) into LDS
// with fully coalesced NON-TEMPORAL global_load_b128 (8 lanes = one 128 B
// row segment), so HBM sees pure wide streams instead of 8 B/512 B-stride
// gathers. WMMA B-fragments come from LDS (ds_load_b64, immediate offsets,
// no address VALU); the U slice (16x128, 8 KB) is staged once per c-block.
// ---------------------------------------------------------------------------
#define OBLK 256   // o columns per block
#define DCH  32    // d-chunk (floats) staged per iteration
__global__ void hyper_w1_kernel(const float* __restrict__ fc1_w,
                                const float* __restrict__ fc1_b,
                                float* __restrict__ ws) {
  __shared__ float uS[R_SZ * D_SZ];     //  8 KB: U[:, c*128 .. c*128+127]
  __shared__ float bS[OBLK * DCH];      // 32 KB: fc1_w chunk
  const int tid = threadIdx.x;          // 256
  const int wave = tid >> 5;
  const int lane = tid & 31;
  const int half = lane >> 4, ln = lane & 15;
  const int obase = blockIdx.x * OBLK;
  v8f acc[2] = {};

  for (int c = 0; c < C_SZ; ++c) {
    const float* gC = fc1_w + (size_t)c * ((size_t)O1 * D_SZ) + (size_t)obase * D_SZ;
    for (int db = 0; db < D_SZ; db += DCH) {
      __syncthreads();   // previous chunk fully consumed
      if (db == 0) {     // stage U slice for this c (512 float4, 2/thread)
#pragma unroll
        for (int i = 0; i < 2; ++i) {
          const int idx = tid + i * 256;                // float4 index
          *(v4f*)(&uS[idx << 2]) =
              *(const v4f*)(ws + OFF_U + (size_t)(idx >> 5) * KU + c * D_SZ +
                            ((idx & 31) << 2));
        }
      }
      // stage B chunk: 256 rows x 32 floats = 2048 float4, 8 per thread;
      // 8 consecutive threads cover one row's 128 B segment (coalesced).
#pragma unroll
      for (int i = 0; i < 8; ++i) {
        const int idx = tid + i * 256;
        const int o_l = idx >> 3;
        const int doff = (idx & 7) << 2;
        const v4f v = __builtin_nontemporal_load(
            (const v4f*)(gC + (size_t)o_l * D_SZ + db + doff));
        *(v4f*)(&bS[o_l * DCH + doff]) = v;
      }
      __syncthreads();
      const float* uRow = &uS[ln * D_SZ + db];
      const float* bRow0 = &bS[(wave * 32 + ln) * DCH];
      const float* bRow1 = &bS[(wave * 32 + 16 + ln) * DCH];
#pragma unroll
      for (int d0 = 0; d0 < DCH; d0 += 4) {
        const int dd = d0 + 2 * half;   // lanes 16-31 supply K+2,K+3
        const v2f a = *(const v2f*)(uRow + dd);
        const v2f b0 = *(const v2f*)(bRow0 + dd);
        const v2f b1 = *(const v2f*)(bRow1 + dd);
        acc[0] = wmma_f32(a, b0, acc[0]);
        acc[1] = wmma_f32(a, b1, acc[1]);
      }
    }
  }
#pragma unroll
  for (int t = 0; t < 2; ++t) {
    const int o = obase + wave * 32 + t * 16 + ln;
    float fb[C_SZ];
#pragma unroll
    for (int c = 0; c < C_SZ; ++c) fb[c] = fc1_b[(size_t)c * O1 + o];
#pragma unroll
    for (int v = 0; v < 8; ++v) {
      const int rr = v + 8 * half;  // C/D layout: VGPR v holds rows v, v+8
      float bias = 0.f;
#pragma unroll
      for (int c = 0; c < C_SZ; ++c) bias += ws[OFF_W + rr * C_SZ + c] * fb[c];
      ws[OFF_W1 + (size_t)rr * O1 + o] = acc[t][v] + bias;
    }
  }
}

// ---------------------------------------------------------------------------
// Kernel 4: feature = x @ base_w.T + base_b. M=4096, N=512, K=1024.
// 4 f-tiles per wave: A fragment reused 4x. x/base_w are L2-resident.
// ---------------------------------------------------------------------------
__global__ void base_feature_kernel(const float* __restrict__ x,
                                    const float* __restrict__ bw,
                                    const float* __restrict__ bb,
                                    float* __restrict__ ws) {
  const int lane = threadIdx.x;
  const int half = lane >> 4, ln = lane & 15;
  const int m = blockIdx.x * 16 + ln;       // batch row (A)
  const int fbase = blockIdx.y * 64;        // four 16-wide feature tiles
  const float* xrow = x + (size_t)m * IN_SZ + 2 * half;
  const float* brow0 = bw + (size_t)(fbase + ln) * IN_SZ + 2 * half;
  v8f acc[4] = {};
#pragma unroll 8
  for (int k0 = 0; k0 < IN_SZ; k0 += 4) {
    const v2f a = *(const v2f*)(xrow + k0);
#pragma unroll
    for (int t = 0; t < 4; ++t) {
      const v2f b = *(const v2f*)(brow0 + (size_t)t * 16 * IN_SZ + k0);
      acc[t] = wmma_f32(a, b, acc[t]);
    }
  }
#pragma unroll
  for (int t = 0; t < 4; ++t) {
    const int f = fbase + t * 16 + ln;
    const float bias = bb[f];
#pragma unroll
    for (int v = 0; v < 8; ++v) {
      const int row = blockIdx.x * 16 + v + 8 * half;
      ws[OFF_FEAT + (size_t)row * F_SZ + f] = acc[t][v] + bias;
    }
  }
}

// ---------------------------------------------------------------------------
// Kernel 5: fused head. 256-thread blocks (8 waves, same ray r, 8 adjacent
// 16-row batch tiles). W1[r] K-chunks (128h x 64k = 32 KB) staged in LDS
// once per block, consumed by all 8 waves via ds_load_b64 (W1 traffic 8x
// lower). v1 stays in registers; +b1, dot with W2 (shfl_xor tree over the
// 16-lane half matching the C/D layout), +b2, sigmoid; writes both outputs
// (sigmoid first, raw second, per the tuple return order).
// ---------------------------------------------------------------------------
#define CH 64   // K-chunk staged in LDS
__global__ void head_kernel(const float* __restrict__ ws, float* __restrict__ out) {
  __shared__ float lds_w1[HID_SZ * CH];       // 32 KB
  const int tid = threadIdx.x;
  const int wave = tid >> 5;
  const int lane = tid & 31;
  const int half = lane >> 4, ln = lane & 15;
  const int r = blockIdx.y;
  const int btile = blockIdx.x * 8 + wave;
  const float* featrow = ws + OFF_FEAT + (size_t)(btile * 16 + ln) * F_SZ;
  const float* W1g = ws + OFF_W1 + (size_t)r * O1;   // [h*512 + f]
  v8f acc[8] = {};

  for (int kbase = 0; kbase < F_SZ; kbase += CH) {
    __syncthreads();   // previous chunk fully consumed
    // Cooperative stage: 128 rows x 64 floats = 2048 float4, 8 per thread.
#pragma unroll
    for (int i = 0; i < 8; ++i) {
      const int idx = tid + i * 256;          // float4 index
      const int h = idx >> 4;
      const int koff = (idx & 15) << 2;
      const v4f v = *(const v4f*)(W1g + (size_t)h * F_SZ + kbase + koff);
      *(v4f*)(&lds_w1[h * CH + koff]) = v;
    }
    __syncthreads();
#pragma unroll 4
    for (int ks = 0; ks < CH; ks += 4) {
      const int kk = ks + 2 * half;
      const v2f a = *(const v2f*)(featrow + kbase + kk);
#pragma unroll
      for (int nt = 0; nt < 8; ++nt) {
        const v2f b = *(const v2f*)(&lds_w1[(nt * 16 + ln) * CH + kk]);
        acc[nt] = wmma_f32(a, b, acc[nt]);
      }
    }
  }

  float bb[8], wwv[8];
#pragma unroll
  for (int nt = 0; nt < 8; ++nt) {
    const int h = nt * 16 + ln;
    bb[nt] = ws[OFF_B1 + r * HID_SZ + h];
    wwv[nt] = ws[OFF_W2 + r * HID_SZ + h];
  }
  const float b2v = ws[OFF_B2 + r];
#pragma unroll
  for (int v = 0; v < 8; ++v) {
    float p = 0.f;
#pragma unroll
    for (int nt = 0; nt < 8; ++nt) p += (acc[nt][v] + bb[nt]) * wwv[nt];
    p += __shfl_xor(p, 1);
    p += __shfl_xor(p, 2);
    p += __shfl_xor(p, 4);
    p += __shfl_xor(p, 8);
    if (ln == 0) {
      const int bidx = btile * 16 + v + 8 * half;
      const float raw = p + b2v;
      const float sig = 1.f / (1.f + __expf(-raw));
      out[(size_t)r * B_SZ + bidx] = sig;                       // sigmoid(out)
      out[(size_t)R_SZ * B_SZ + (size_t)r * B_SZ + bidx] = raw; // out
    }
  }
}

// ---------------------------------------------------------------------------
extern "C" void kernel_launch(void* const* d_in, const int* in_sizes, int n_in,
                              void* d_out, int out_size, void* d_ws, size_t ws_size,
                              hipStream_t stream) {
  const float* x      = (const float*)d_in[0];
  const float* pref   = (const float*)d_in[1];
  const float* base_w = (const float*)d_in[2];
  const float* base_b = (const float*)d_in[3];
  const float* wm1_w  = (const float*)d_in[4];
  const float* wm1_b  = (const float*)d_in[5];
  const float* wm2_w  = (const float*)d_in[6];
  const float* wm2_b  = (const float*)d_in[7];
  const float* ray0_w = (const float*)d_in[8];
  const float* ray0_b = (const float*)d_in[9];
  const float* ray2_w = (const float*)d_in[10];
  const float* ray2_b = (const float*)d_in[11];
  const float* fc1_w  = (const float*)d_in[12];
  const float* fc1_b  = (const float*)d_in[13];
  const float* b1_w   = (const float*)d_in[14];
  const float* b1_b   = (const float*)d_in[15];
  const float* fc2_w  = (const float*)d_in[16];
  const float* fc2_b  = (const float*)d_in[17];
  const float* b2_w   = (const float*)d_in[18];
  const float* b2_b   = (const float*)d_in[19];
  (void)in_sizes; (void)n_in; (void)out_size; (void)ws_size;

  float* ws  = (float*)d_ws;   // needs ~12.7 MB
  float* out = (float*)d_out;

  prep_kernel<<<R_SZ, 128, 0, stream>>>(pref, wm1_w, wm1_b, wm2_w, wm2_b,
                                        ray0_w, ray0_b, ray2_w, ray2_b, ws);
  small_hyper_kernel<<<R_SZ, 128, 0, stream>>>(b1_w, b1_b, ws, ws + OFF_B1, HID_SZ);
  small_hyper_kernel<<<R_SZ, 128, 0, stream>>>(fc2_w, fc2_b, ws, ws + OFF_W2, HID_SZ);
  small_hyper_kernel<<<R_SZ, 32, 0, stream>>>(b2_w, b2_b, ws, ws + OFF_B2, OUT_SZ);
  hyper_w1_kernel<<<O1 / OBLK, 256, 0, stream>>>(fc1_w, fc1_b, ws);
  base_feature_kernel<<<dim3(B_SZ / 16, F_SZ / 64), 32, 0, stream>>>(x, base_w, base_b, ws);
  head_kernel<<<dim3(B_SZ / 128, R_SZ), 256, 0, stream>>>(ws, out);
}